// SM2_42820823941688
// MI455X (gfx1250) — compile-verified
//
#include <hip/hip_runtime.h>
#include <hip/hip_bf16.h>
#include <stdint.h>

// SM-2 scan: inputs (512, 32768, 2) f32, w (6) f32
// outputs: (512, 32768, 3) f32 then last_state (32768, 3) f32, concatenated.

#define SEQ_LEN 512
#define BATCH   32768
#define TPB     256
#define CH      16                 // time steps per async chunk
#define NCH     (SEQ_LEN / CH)     // 32 chunks

struct __align__(4) F3 { float x, y, z; };

__device__ __forceinline__ uint32_t lds_off(const void* p) {
    // generic pointer to __shared__: low 32 bits are the 0-based LDS byte address
    return (uint32_t)(uintptr_t)p;
}

__device__ __forceinline__ void async_load_b64(uint32_t lds_addr, const void* gaddr) {
    asm volatile("global_load_async_to_lds_b64 %0, %1, off"
                 :: "v"(lds_addr), "v"(gaddr) : "memory");
}

__global__ __launch_bounds__(TPB) void sm2_scan_async_kernel(
    const float* __restrict__ in,   // (SEQ_LEN, BATCH, 2)
    const float* __restrict__ w6,   // 6 params
    float* __restrict__ out)        // (SEQ_LEN, BATCH, 3) then (BATCH, 3)
{
    __shared__ float2 buf[2][CH][TPB];   // 64 KB, double-buffered input stage

    const int tid = threadIdx.x;
    const int col = blockIdx.x * TPB + tid;

    const float w0 = w6[0], w1 = w6[1], w2 = w6[2];
    const float w3 = w6[3], w4 = w6[4], w5 = w6[5];

    // this thread's column, float2 (8B) units; step t lives at gcol + t*BATCH
    const float2* gcol = (const float2*)in + col;

    // prime the pipeline: issue chunks 0 and 1 (32 async loads / wave in flight)
    #pragma unroll
    for (int s = 0; s < CH; ++s)
        async_load_b64(lds_off(&buf[0][s][tid]), gcol + (size_t)s * BATCH);
    #pragma unroll
    for (int s = 0; s < CH; ++s)
        async_load_b64(lds_off(&buf[1][s][tid]), gcol + (size_t)(CH + s) * BATCH);

    float ivl  = 0.0f;
    float ef   = w2;
    float reps = 0.0f;

    for (int c = 0; c < NCH; ++c) {
        // oldest chunk must be complete; loads return done in order per wave
        if (c < NCH - 1)
            asm volatile("s_wait_asynccnt 16" ::: "memory");
        else
            asm volatile("s_wait_asynccnt 0" ::: "memory");

        const int bsel = c & 1;
        const int t0   = c * CH;

        #pragma unroll
        for (int s = 0; s < CH; ++s) {
            float2 v = buf[bsel][s][tid];   // ds_load_b64, conflict-free
            float rating = v.y;

            // new_ef = clip(ef - w3*(rating+1 - w4)^2 + w5, 1.3, 10.0)
            float d    = (rating + 1.0f) - w4;
            float ef_n = __builtin_fmaf(-w3, d * d, ef) + w5;
            ef = fminf(fmaxf(ef_n, 1.3f), 10.0f);

            // new_reps = success ? reps+1 : 0
            reps = (rating > 1.0f) ? (reps + 1.0f) : 0.0f;

            // new_ivl: reps 0/1 -> w0, reps 2 -> w1, else ivl * new_ef
            float ivl_n = ivl * ef;
            ivl_n = (reps < 2.5f) ? w1 : ivl_n;
            ivl_n = (reps < 1.5f) ? w0 : ivl_n;
            ivl = fminf(fmaxf(ivl_n, 0.01f), 36500.0f);

            // outputs[t][col] = {ivl, ef, reps}: contiguous b96 per lane
            F3 val; val.x = ivl; val.y = ef; val.z = reps;
            *(F3*)(out + ((size_t)(t0 + s) * BATCH + col) * 3) = val;
        }

        // refill the buffer we just drained (all its ds_loads are consumed above)
        if (c + 2 < NCH) {
            #pragma unroll
            for (int s = 0; s < CH; ++s)
                async_load_b64(lds_off(&buf[bsel][s][tid]),
                               gcol + (size_t)((c + 2) * CH + s) * BATCH);
        }
    }

    // last_state = state after step 511
    F3 val; val.x = ivl; val.y = ef; val.z = reps;
    *(F3*)(out + (size_t)SEQ_LEN * BATCH * 3 + (size_t)col * 3) = val;
}

extern "C" void kernel_launch(void* const* d_in, const int* in_sizes, int n_in,
                              void* d_out, int out_size, void* d_ws, size_t ws_size,
                              hipStream_t stream) {
    const float* in = (const float*)d_in[0];   // (512, 32768, 2) f32
    const float* w  = (const float*)d_in[1];   // (6,) f32
    float* out = (float*)d_out;

    dim3 grid(BATCH / TPB);   // 128 blocks x 256 threads = 1 thread per column
    sm2_scan_async_kernel<<<grid, TPB, 0, stream>>>(in, w, out);
}